// GATLayer_32263794328265
// MI455X (gfx1250) — compile-verified
//
#include <hip/hip_runtime.h>
#include <hip/hip_bf16.h>
#include <math.h>

// GAT layer for MI455X (gfx1250, wave32).
//
// Phase 1: xp = x @ W^T      -> V_WMMA_F32_16X16X4_F32 (exact fp32 matrix op)
// Phase 2: s,d per-node logits (tiny dot products)
// Phase 3: edge logit segment-max (float atomic max via int min/max trick)
// Phase 4: segment-sum of exp   (global_atomic_add_f32)
// Phase 5: weighted scatter-aggregate (gather xp[col], atomicAdd into out[row])

#define IN_DIM   128
#define NUM_HEADS 8
#define HEAD_DIM 16
#define NEG_SLOPE 0.2f

typedef __attribute__((ext_vector_type(2))) float v2f;
typedef __attribute__((ext_vector_type(8))) float v8f;

// ---------------------------------------------------------------------------
// Phase 1: WMMA projection. One wave per 16-row M strip, covering all 8 N
// tiles (N = 128). K = 128 in steps of 4 (f32 WMMA shape 16x16x4).
//
// f32 16x4 A fragment layout (ISA 7.12.2): lane<16 holds M=lane, K={k0,k0+1}
// in VGPR0/1; lane>=16 holds M=lane-16, K={k0+2,k0+3}. That is one float2 per
// lane at x[(row)*128 + k0 + 2*(lane>>4)]. B = W^T has B[k][n] = W[n][k], so
// the mirrored B fragment is a float2 at W[n*128 + k0 + 2*(lane>>4)] -- both
// are contiguous, 8B-aligned vector loads.
// ---------------------------------------------------------------------------
__global__ void __launch_bounds__(128)
gat_proj_wmma(const float* __restrict__ x, const float* __restrict__ W,
              float* __restrict__ xp, int n_tiles) {
  const int wave = threadIdx.x >> 5;
  const int lane = threadIdx.x & 31;
  const int tile = blockIdx.x * 4 + wave;
  if (tile >= n_tiles) return;          // whole wave exits: EXEC stays all-1s

  const int row_base = tile * 16;
  const int mrow = lane & 15;
  const int kgrp = (lane >> 4) * 2;     // 0 or 2

  v8f acc[8];
#pragma unroll
  for (int i = 0; i < 8; ++i) acc[i] = (v8f){0.f,0.f,0.f,0.f,0.f,0.f,0.f,0.f};

  const float* arow = x + (size_t)(row_base + mrow) * IN_DIM + kgrp;

#pragma unroll 4
  for (int k0 = 0; k0 < IN_DIM; k0 += 4) {
    const v2f a = *(const v2f*)(arow + k0);
#pragma unroll
    for (int nt = 0; nt < 8; ++nt) {
      const v2f b = *(const v2f*)(W + (size_t)(nt * 16 + mrow) * IN_DIM + k0 + kgrp);
      // D = A x B + C  (8 args: neg_a, A, neg_b, B, c_mod, C, reuse_a, reuse_b)
      acc[nt] = __builtin_amdgcn_wmma_f32_16x16x4_f32(
          false, a, false, b, (short)0, acc[nt], false, false);
    }
  }

  // C/D layout: VGPR r -> M = r + (lane>=16 ? 8 : 0), N = lane&15.
  const int mo = (lane >> 4) * 8;
#pragma unroll
  for (int nt = 0; nt < 8; ++nt) {
#pragma unroll
    for (int r = 0; r < 8; ++r) {
      xp[(size_t)(row_base + mo + r) * IN_DIM + nt * 16 + (lane & 15)] = acc[nt][r];
    }
  }
}

// ---------------------------------------------------------------------------
// Phase 2: per-node attention logits  s[n,h] = <xp[n,h,:], src_attn[h,:]>
// ---------------------------------------------------------------------------
__global__ void __launch_bounds__(256)
gat_node_logits(const float* __restrict__ xp,
                const float* __restrict__ src_attn,
                const float* __restrict__ dst_attn,
                float* __restrict__ s, float* __restrict__ d, int n_nodes) {
  int gid = blockIdx.x * blockDim.x + threadIdx.x;
  if (gid >= n_nodes * NUM_HEADS) return;
  int node = gid / NUM_HEADS;
  int h = gid % NUM_HEADS;
  const float* v = xp + (size_t)node * IN_DIM + h * HEAD_DIM;
  float ss = 0.f, dd = 0.f;
#pragma unroll
  for (int k = 0; k < HEAD_DIM; ++k) {
    float xv = v[k];
    ss += xv * src_attn[h * HEAD_DIM + k];
    dd += xv * dst_attn[h * HEAD_DIM + k];
  }
  s[gid] = ss;
  d[gid] = dd;
}

// ---------------------------------------------------------------------------
// Init: out = 0, m = -inf, denom = 0 (fresh every launch; harness poisons).
// ---------------------------------------------------------------------------
__global__ void __launch_bounds__(256)
gat_init(float* __restrict__ out, float* __restrict__ m,
         float* __restrict__ denom, int n_nodes) {
  int gid = blockIdx.x * blockDim.x + threadIdx.x;
  if (gid < n_nodes * IN_DIM) out[gid] = 0.f;
  if (gid < n_nodes * NUM_HEADS) {
    m[gid] = -INFINITY;
    denom[gid] = 0.f;
  }
}

// float atomic-max via signed max / unsigned min (lowers to native i32/u32
// global atomics; monotonic-correct for the IEEE-754 total order on floats).
__device__ __forceinline__ void atomicMaxFloat(float* addr, float val) {
  if (val >= 0.f) atomicMax((int*)addr, __float_as_int(val));
  else            atomicMin((unsigned int*)addr, (unsigned int)__float_as_int(val));
}

__device__ __forceinline__ float leaky(float v) {
  return v >= 0.f ? v : NEG_SLOPE * v;
}

// ---------------------------------------------------------------------------
// Phase 3: segment max of edge logits. One thread per (edge, head).
// ---------------------------------------------------------------------------
__global__ void __launch_bounds__(256)
gat_edge_max(const long long* __restrict__ rows, const long long* __restrict__ cols,
             const float* __restrict__ s, const float* __restrict__ d,
             float* __restrict__ m, int n_edges) {
  int gid = blockIdx.x * blockDim.x + threadIdx.x;
  if (gid >= n_edges * NUM_HEADS) return;
  int e = gid / NUM_HEADS;
  int h = gid % NUM_HEADS;
  long long r = rows[e], c = cols[e];
  float v = leaky(s[r * NUM_HEADS + h] + d[c * NUM_HEADS + h]);
  atomicMaxFloat(&m[r * NUM_HEADS + h], v);
}

// ---------------------------------------------------------------------------
// Phase 4: denom[row,h] += exp(e - m[row,h]). One thread per (edge, head).
// ---------------------------------------------------------------------------
__global__ void __launch_bounds__(256)
gat_edge_sum(const long long* __restrict__ rows, const long long* __restrict__ cols,
             const float* __restrict__ s, const float* __restrict__ d,
             const float* __restrict__ m, float* __restrict__ denom, int n_edges) {
  int gid = blockIdx.x * blockDim.x + threadIdx.x;
  if (gid >= n_edges * NUM_HEADS) return;
  int e = gid / NUM_HEADS;
  int h = gid % NUM_HEADS;
  long long r = rows[e], c = cols[e];
  float v = leaky(s[r * NUM_HEADS + h] + d[c * NUM_HEADS + h]);
  atomicAdd(&denom[r * NUM_HEADS + h], __expf(v - m[r * NUM_HEADS + h]));
}

// ---------------------------------------------------------------------------
// Phase 5: out[row, ch] += (exp(e-m)/denom) * xp[col, ch].
// One thread per (edge, channel). Recompute the softmax weight on the fly
// (s/d/m/denom are tiny and L2-hot) instead of storing a 51 MB [E,H] buffer.
// ---------------------------------------------------------------------------
__global__ void __launch_bounds__(256)
gat_aggregate(const long long* __restrict__ rows, const long long* __restrict__ cols,
              const float* __restrict__ s, const float* __restrict__ d,
              const float* __restrict__ m, const float* __restrict__ denom,
              const float* __restrict__ xp, float* __restrict__ out, int n_edges) {
  long long gid = (long long)blockIdx.x * blockDim.x + threadIdx.x;
  if (gid >= (long long)n_edges * IN_DIM) return;
  int e  = (int)(gid >> 7);   // / 128
  int ch = (int)(gid & 127);
  int h  = ch >> 4;           // / HEAD_DIM
  long long r = rows[e], c = cols[e];
  float v = leaky(s[r * NUM_HEADS + h] + d[c * NUM_HEADS + h]);
  float w = __expf(v - m[r * NUM_HEADS + h]) / denom[r * NUM_HEADS + h];
  atomicAdd(&out[r * IN_DIM + ch], w * xp[c * IN_DIM + ch]);
}

// ---------------------------------------------------------------------------
extern "C" void kernel_launch(void* const* d_in, const int* in_sizes, int n_in,
                              void* d_out, int out_size, void* d_ws, size_t ws_size,
                              hipStream_t stream) {
  const float*     x        = (const float*)d_in[0];
  const long long* ei       = (const long long*)d_in[1];   // int64 per reference
  const float*     W        = (const float*)d_in[2];
  const float*     src_attn = (const float*)d_in[3];
  const float*     dst_attn = (const float*)d_in[4];
  float*           out      = (float*)d_out;

  const int n_nodes = in_sizes[0] / IN_DIM;       // 100000
  const int n_edges = in_sizes[1] / 2;            // 1600000
  const long long* rows = ei;
  const long long* cols = ei + n_edges;

  // Workspace layout: xp[N*128] | s[N*8] | d[N*8] | m[N*8] | denom[N*8]
  float* xp    = (float*)d_ws;
  float* s     = xp    + (size_t)n_nodes * IN_DIM;
  float* dl    = s     + (size_t)n_nodes * NUM_HEADS;
  float* m     = dl    + (size_t)n_nodes * NUM_HEADS;
  float* denom = m     + (size_t)n_nodes * NUM_HEADS;

  // Phase 1: WMMA GEMM. 16-row tiles; 100000 = 6250 * 16 exactly.
  const int n_tiles = n_nodes / 16;
  dim3 gemm_grid((n_tiles + 3) / 4);
  gat_proj_wmma<<<gemm_grid, 128, 0, stream>>>(x, W, xp, n_tiles);

  // Phase 2: node logits.
  int nh = n_nodes * NUM_HEADS;
  gat_node_logits<<<(nh + 255) / 256, 256, 0, stream>>>(xp, src_attn, dst_attn,
                                                        s, dl, n_nodes);

  // Init accumulators + output.
  int nc = n_nodes * IN_DIM;
  gat_init<<<(nc + 255) / 256, 256, 0, stream>>>(out, m, denom, n_nodes);

  // Phase 3/4: segment softmax statistics.
  int eh = n_edges * NUM_HEADS;
  gat_edge_max<<<(eh + 255) / 256, 256, 0, stream>>>(rows, cols, s, dl, m, n_edges);
  gat_edge_sum<<<(eh + 255) / 256, 256, 0, stream>>>(rows, cols, s, dl, m, denom,
                                                     n_edges);

  // Phase 5: scatter-aggregate.
  long long ec = (long long)n_edges * IN_DIM;
  int agg_blocks = (int)((ec + 255) / 256);
  gat_aggregate<<<agg_blocks, 256, 0, stream>>>(rows, cols, s, dl, m, denom,
                                                xp, out, n_edges);
}